// rgnn_66563403153454
// MI455X (gfx1250) — compile-verified
//
#include <hip/hip_runtime.h>

#define NODES     62
#define NPAD      64                  // padded per-graph row count for H2
#define BATCH     512
#define NTOT      (NODES * BATCH)     // 31744 (unpadded x rows)
#define NUM_IN    128
#define NUM_HID   256
#define NUM_CLASS 3

typedef __attribute__((ext_vector_type(2))) float v2f;
typedef __attribute__((ext_vector_type(8))) float v8f;

// ---------------------------------------------------------------------------
// Kernel A: build the 2-hop propagation matrix M = (P^T)^2, zero-padded 64x64.
//   P[r,c] = dis[r]*W[r,c]*dis[c] + (r==c)*dis[c]^2
//   Q = P^T, M = Q*Q   (so h2 = M @ h applies two hops)
// Rows/cols >= 62 of M are exactly zero -> padded K reads and padded output
// rows are mathematically inert.
// ---------------------------------------------------------------------------
__global__ void rgnn_build_M(const float* __restrict__ w, float* __restrict__ M) {
    __shared__ float sDis[64];
    __shared__ float sQ[64 * 64];
    int tid = threadIdx.x;

    if (tid < 64) {
        float dis = 0.f;
        if (tid < NODES) {
            float deg = 1.f;                          // self-loop weight 1.0
            for (int r = 0; r < NODES; ++r) deg += w[r * NODES + tid];
            dis = (deg > 0.f) ? rsqrtf(deg) : 0.f;
        }
        sDis[tid] = dis;
    }
    __syncthreads();

    // Q[i][j] = P[j][i]
    for (int idx = tid; idx < 64 * 64; idx += blockDim.x) {
        int i = idx >> 6, j = idx & 63;
        float q = 0.f;
        if (i < NODES && j < NODES) {
            q = sDis[j] * w[j * NODES + i] * sDis[i];
            if (i == j) q += sDis[i] * sDis[i];
        }
        sQ[idx] = q;
    }
    __syncthreads();

    for (int idx = tid; idx < 64 * 64; idx += blockDim.x) {
        int i = idx >> 6, j = idx & 63;
        float acc = 0.f;
        for (int k = 0; k < 64; ++k) acc += sQ[i * 64 + k] * sQ[k * 64 + j];
        M[idx] = acc;
    }
}

// ---------------------------------------------------------------------------
// Kernel B: per graph, H2 = M @ X_g  -> padded [64 x 128] per graph.
// One block per graph, 8 waves. Wave w: rowtile rt = w>>1, col half cg = w&1.
// fp32 WMMA 16x16x4, K loop over 64 in steps of 4. Unconditional stores:
// output rows 62/63 are exact zeros (M rows >= 62 are zero).
// ---------------------------------------------------------------------------
__global__ __launch_bounds__(256) void rgnn_prop(const float* __restrict__ M,
                                                 const float* __restrict__ x,
                                                 float* __restrict__ h2) {
    int g    = blockIdx.x;
    int tid  = threadIdx.x;
    int lane = tid & 31;
    int wave = tid >> 5;          // 0..7
    int l15  = lane & 15;
    int hi   = lane >> 4;         // 0 or 1
    int khalf = hi << 1;          // 0 or 2  (A/B K sub-offset per ISA layout)
    int rt   = wave >> 1;         // 0..3
    int cg   = wave & 1;          // 0..1

    int arow = rt * 16 + l15;     // output local row (row of M)
    v8f acc[4];
    for (int t = 0; t < 4; ++t)
        acc[t] = (v8f){0.f, 0.f, 0.f, 0.f, 0.f, 0.f, 0.f, 0.f};

    int baseRow = g * NODES;      // x is unpadded (62 rows/graph)
    for (int k0 = 0; k0 < 64; k0 += 4) {
        // A pair is contiguous & 8B-aligned -> single b64 load
        v2f a = *(const v2f*)&M[arow * 64 + k0 + khalf];
        int r0 = baseRow + k0 + khalf;     if (r0 > NTOT - 1) r0 = NTOT - 1;
        int r1 = baseRow + k0 + khalf + 1; if (r1 > NTOT - 1) r1 = NTOT - 1;
        // K >= 62 reads hit neighbor/clamped rows but multiply zero M columns.
        for (int t = 0; t < 4; ++t) {
            int col = (cg * 4 + t) * 16 + l15;
            v2f b;
            b.x = x[r0 * NUM_IN + col];
            b.y = x[r1 * NUM_IN + col];
            acc[t] = __builtin_amdgcn_wmma_f32_16x16x4_f32(
                false, a, false, b, (short)0, acc[t], false, false);
        }
    }
    int gbase = g * NPAD;         // padded H2 layout
    for (int t = 0; t < 4; ++t) {
        int col = (cg * 4 + t) * 16 + l15;
        for (int v = 0; v < 8; ++v) {
            int rloc = rt * 16 + v + hi * 8;   // D layout: VGPR v -> row v (+8 hi lanes)
            h2[(gbase + rloc) * NUM_IN + col] = acc[t][v];
        }
    }
}

// ---------------------------------------------------------------------------
// Kernel C: per graph, Y = H2 @ lin_w^T + lin_b  (64x128 x 128x256), then
// row-softmax over 256 (62 valid rows), sum-pool, FC 256->3, softmax over 3.
// One block per graph, 16 waves. Wave w: rowtile rt = w>>2, 64-col group cg = w&3.
// ---------------------------------------------------------------------------
__global__ __launch_bounds__(512) void rgnn_head(const float* __restrict__ h2,
                                                 const float* __restrict__ lin_w,
                                                 const float* __restrict__ lin_b,
                                                 const float* __restrict__ fc_w,
                                                 const float* __restrict__ fc_b,
                                                 float* __restrict__ out) {
    __shared__ float sY[NODES * NUM_HID];   // 62*256 fp32 = 63488 B
    __shared__ float sMax[64];
    __shared__ float sInvZ[64];
    __shared__ float sPool[NUM_HID];
    __shared__ float sLogit[4];

    int g    = blockIdx.x;
    int tid  = threadIdx.x;
    int lane = tid & 31;
    int wave = tid >> 5;          // 0..15
    int l15  = lane & 15;
    int hi   = lane >> 4;
    int khalf = hi << 1;
    int rt   = wave >> 2;         // 0..3
    int cg   = wave & 3;          // 0..3 -> columns [cg*64, cg*64+64)

    int rA = g * NPAD + rt * 16 + l15;      // padded H2: no clamp needed

    v8f acc[4];
    for (int t = 0; t < 4; ++t)
        acc[t] = (v8f){0.f, 0.f, 0.f, 0.f, 0.f, 0.f, 0.f, 0.f};

    for (int k0 = 0; k0 < NUM_IN; k0 += 4) {
        v2f a = *(const v2f*)&h2[rA * NUM_IN + k0 + khalf];
        for (int t = 0; t < 4; ++t) {
            int col = cg * 64 + t * 16 + l15;        // output channel
            // B[k][n] = lin_w[n][k]; (k,k+1) contiguous in lin_w's row
            v2f b = *(const v2f*)&lin_w[col * NUM_IN + k0 + khalf];
            acc[t] = __builtin_amdgcn_wmma_f32_16x16x4_f32(
                false, a, false, b, (short)0, acc[t], false, false);
        }
    }
    for (int t = 0; t < 4; ++t) {
        int col = cg * 64 + t * 16 + l15;
        float bias = lin_b[col];
        for (int v = 0; v < 8; ++v) {
            int rloc = rt * 16 + v + hi * 8;
            if (rloc < NODES) sY[rloc * NUM_HID + col] = acc[t][v] + bias;
        }
    }
    __syncthreads();

    // per-row softmax statistics (62 rows; one thread per row)
    if (tid < NODES) {
        float m = -3.402823466e38f;
        for (int c = 0; c < NUM_HID; ++c) m = fmaxf(m, sY[tid * NUM_HID + c]);
        float z = 0.f;
        for (int c = 0; c < NUM_HID; ++c) z += expf(sY[tid * NUM_HID + c] - m);
        sMax[tid]  = m;
        sInvZ[tid] = 1.f / z;                       // z >= 1 always
    }
    __syncthreads();

    // global_add_pool of the row-softmaxed activations
    if (tid < NUM_HID) {
        float p = 0.f;
        for (int r = 0; r < NODES; ++r)
            p += expf(sY[r * NUM_HID + tid] - sMax[r]) * sInvZ[r];
        sPool[tid] = p;
    }
    __syncthreads();

    // FC 256 -> 3
    if (tid < NUM_CLASS) {
        float a0 = fc_b[tid];
        for (int o = 0; o < NUM_HID; ++o) a0 += sPool[o] * fc_w[tid * NUM_HID + o];
        sLogit[tid] = a0;
    }
    __syncthreads();

    // final softmax over 3 classes
    if (tid < NUM_CLASS) {
        float l0 = sLogit[0], l1 = sLogit[1], l2 = sLogit[2];
        float m  = fmaxf(l0, fmaxf(l1, l2));
        float e0 = expf(l0 - m), e1 = expf(l1 - m), e2 = expf(l2 - m);
        float inv = 1.f / (e0 + e1 + e2);
        float mine = (tid == 0) ? e0 : (tid == 1) ? e1 : e2;
        out[g * NUM_CLASS + tid] = mine * inv;
    }
}

// ---------------------------------------------------------------------------
extern "C" void kernel_launch(void* const* d_in, const int* in_sizes, int n_in,
                              void* d_out, int out_size, void* d_ws, size_t ws_size,
                              hipStream_t stream) {
    // setup_inputs order: len_y, x, index, batch, weight, lin_w, lin_b, fc_w, fc_b
    const float* x     = (const float*)d_in[1];
    const float* w     = (const float*)d_in[4];
    const float* lin_w = (const float*)d_in[5];
    const float* lin_b = (const float*)d_in[6];
    const float* fc_w  = (const float*)d_in[7];
    const float* fc_b  = (const float*)d_in[8];
    float* out = (float*)d_out;

    float* M  = (float*)d_ws;          // 64*64 fp32
    float* h2 = M + 64 * 64;           // BATCH*64*128 fp32 (~16.8 MB, padded)

    rgnn_build_M<<<1, 256, 0, stream>>>(w, M);
    rgnn_prop  <<<BATCH, 256, 0, stream>>>(M, x, h2);
    rgnn_head  <<<BATCH, 512, 0, stream>>>(h2, lin_w, lin_b, fc_w, fc_b, out);
}